// GCNLayer_15195594293520
// MI455X (gfx1250) — compile-verified
//
#include <hip/hip_runtime.h>
#include <hip/hip_bf16.h>
#include <stddef.h>

// ---------------------------------------------------------------------------
// GCN layer for MI455X (gfx1250, wave32):
//   agg = segment_sum(x[src] * val)          (COO SpMM, L2-resident scatter)
//   h   = agg @ W + bias                     (V_WMMA_F32_16X16X4_F32, full fp32)
//   out = BN(h) * gamma + beta               (deterministic 2-level reduction)
// ---------------------------------------------------------------------------

#define GCN_N      100000
#define GCN_D      128
#define BN_EPS_F   1e-5f
#define LDSROW     132            // 132 % 64 = 4 banks stride -> conflict-free
#define STAT_ROWS  256

typedef __attribute__((ext_vector_type(2))) float v2f;
typedef __attribute__((ext_vector_type(8))) float v8f;

// ---------------------------------------------------------------------------
// SpMM scatter: one wave (32 lanes) per edge; each lane owns a float4 slice
// (32 lanes * 4 floats = 128 = D). Gather of x[src] is a fully coalesced
// 512B read (L2-resident); scatter uses hardware f32 atomics.
// ---------------------------------------------------------------------------
__global__ void gcn_spmm_kernel(const float* __restrict__ x,
                                const int* __restrict__ esrc,
                                const int* __restrict__ edst,
                                const float* __restrict__ eval,
                                float* __restrict__ agg,
                                int nedges) {
    int e    = blockIdx.x * 8 + (threadIdx.x >> 5);   // 8 waves / block
    int lane = threadIdx.x & 31;
    if (e >= nedges) return;
    int   s = esrc[e];
    int   d = edst[e];
    float v = eval[e];
    const float4* xr = (const float4*)(x + (size_t)s * GCN_D);
    float4 m = xr[lane];
    float* ar = agg + (size_t)d * GCN_D + lane * 4;
    unsafeAtomicAdd(ar + 0, m.x * v);
    unsafeAtomicAdd(ar + 1, m.y * v);
    unsafeAtomicAdd(ar + 2, m.z * v);
    unsafeAtomicAdd(ar + 3, m.w * v);
}

// ---------------------------------------------------------------------------
// GEMM + bias with V_WMMA_F32_16X16X4_F32.
// Block = 256 threads = 8 waves. Block b owns rows [16b, 16b+16); wave w owns
// output columns [16w, 16w+16). A-tile (16x128 f32, 8.25KB padded) staged in
// LDS once and reused by all 8 waves. K = 128 -> 32 WMMA issues per wave.
//
// f32 WMMA operand layout (ISA 7.12.2):
//   A 16x4 : lanes 0-15 M=0..15 {VGPR0=K0, VGPR1=K1}; lanes 16-31 {K2, K3}
//   B 4x16 : VGPR0 {lanes0-15: K0,N=lane | lanes16-31: K2}; VGPR1 {K1 | K3}
//   C/D    : VGPR r -> M = r + 8*(lane>>4), N = lane&15
// ---------------------------------------------------------------------------
__global__ void gcn_gemm_bias_kernel(const float* __restrict__ agg,
                                     const float* __restrict__ W,
                                     const float* __restrict__ bias,
                                     float* __restrict__ out) {
    __shared__ float ldsA[16 * LDSROW];
    const int tid  = threadIdx.x;
    const int row0 = blockIdx.x * 16;

    // Stage A tile: 16 rows x 128 cols, float4-vectorized, padded pitch.
    for (int i = tid; i < 16 * 32; i += 256) {
        int r  = i >> 5;
        int c4 = (i & 31) * 4;
        float4 v = *(const float4*)(agg + (size_t)(row0 + r) * GCN_D + c4);
        ldsA[r * LDSROW + c4 + 0] = v.x;
        ldsA[r * LDSROW + c4 + 1] = v.y;
        ldsA[r * LDSROW + c4 + 2] = v.z;
        ldsA[r * LDSROW + c4 + 3] = v.w;
    }
    __syncthreads();

    const int wave  = tid >> 5;
    const int lane  = tid & 31;
    const int l16   = lane & 15;        // A row (M) / B col (N) within tile
    const int half  = lane >> 4;        // K sub-pair select
    const int col   = wave * 16 + l16;  // global output column

    v8f c = {};
#pragma unroll
    for (int k = 0; k < 32; ++k) {
        const int kb = k * 4 + half * 2;
        v2f a, b;
        a.x = ldsA[l16 * LDSROW + kb];
        a.y = ldsA[l16 * LDSROW + kb + 1];
        b.x = W[(size_t)(kb)     * GCN_D + col];
        b.y = W[(size_t)(kb + 1) * GCN_D + col];
        c = __builtin_amdgcn_wmma_f32_16x16x4_f32(
                /*neg_a=*/false, a, /*neg_b=*/false, b,
                /*c_mod=*/(short)0, c, /*reuse_a=*/false, /*reuse_b=*/false);
    }

    const float bv = bias[col];
#pragma unroll
    for (int r = 0; r < 8; ++r) {
        const int row = row0 + half * 8 + r;
        out[(size_t)row * GCN_D + col] = c[r] + bv;
    }
}

// ---------------------------------------------------------------------------
// BN pass 1: per-block column partial sums (deterministic, no atomics).
// One thread per column; each block covers STAT_ROWS rows.
// ---------------------------------------------------------------------------
__global__ void gcn_colstats_kernel(const float* __restrict__ h,
                                    float* __restrict__ psum,
                                    float* __restrict__ psumsq,
                                    int nrows) {
    const int c    = threadIdx.x;
    const int row0 = blockIdx.x * STAT_ROWS;
    const int rend = min(row0 + STAT_ROWS, nrows);
    float s = 0.0f, s2 = 0.0f;
    for (int row = row0; row < rend; ++row) {
        float v = h[(size_t)row * GCN_D + c];
        s  += v;
        s2 += v * v;
    }
    psum[(size_t)blockIdx.x * GCN_D + c]   = s;
    psumsq[(size_t)blockIdx.x * GCN_D + c] = s2;
}

// ---------------------------------------------------------------------------
// BN pass 2: fold partials, produce per-column scale/shift.
// ---------------------------------------------------------------------------
__global__ void gcn_bnfinalize_kernel(const float* __restrict__ psum,
                                      const float* __restrict__ psumsq,
                                      const float* __restrict__ gamma,
                                      const float* __restrict__ beta,
                                      float* __restrict__ scale,
                                      float* __restrict__ shift,
                                      int nblk, float inv_n) {
    const int c = threadIdx.x;
    float s = 0.0f, s2 = 0.0f;
    for (int b = 0; b < nblk; ++b) {
        s  += psum[(size_t)b * GCN_D + c];
        s2 += psumsq[(size_t)b * GCN_D + c];
    }
    const float mean = s * inv_n;
    const float var  = s2 * inv_n - mean * mean;
    const float sc   = gamma[c] * rsqrtf(var + BN_EPS_F);
    scale[c] = sc;
    shift[c] = beta[c] - mean * sc;
}

// ---------------------------------------------------------------------------
// BN pass 3: in-place affine apply on d_out (h), float4 grid-stride.
// ---------------------------------------------------------------------------
__global__ void gcn_bnapply_kernel(float* __restrict__ out,
                                   const float* __restrict__ scale,
                                   const float* __restrict__ shift,
                                   long nquads) {
    float4* o4 = (float4*)out;
    for (long i = blockIdx.x * (long)blockDim.x + threadIdx.x; i < nquads;
         i += (long)gridDim.x * blockDim.x) {
        const int cb = (int)(i & 31) * 4;   // column of first lane of the quad
        float4 v = o4[i];
        v.x = v.x * scale[cb + 0] + shift[cb + 0];
        v.y = v.y * scale[cb + 1] + shift[cb + 1];
        v.z = v.z * scale[cb + 2] + shift[cb + 2];
        v.w = v.w * scale[cb + 3] + shift[cb + 3];
        o4[i] = v;
    }
}

// ---------------------------------------------------------------------------
// Host launcher (graph-capture safe: only kernel launches + hipMemsetAsync).
// ---------------------------------------------------------------------------
extern "C" void kernel_launch(void* const* d_in, const int* in_sizes, int n_in,
                              void* d_out, int out_size, void* d_ws, size_t ws_size,
                              hipStream_t stream) {
    const float* x     = (const float*)d_in[0];
    const int*   esrc  = (const int*)d_in[1];
    const int*   edst  = (const int*)d_in[2];
    const float* eval  = (const float*)d_in[3];
    const float* W     = (const float*)d_in[4];
    const float* bias  = (const float*)d_in[5];
    const float* gamma = (const float*)d_in[6];
    const float* beta  = (const float*)d_in[7];
    float*       out   = (float*)d_out;

    const int E = in_sizes[1];
    const int N = in_sizes[0] / GCN_D;          // 100000
    const int n_stat_blk = (N + STAT_ROWS - 1) / STAT_ROWS;

    // Workspace layout
    float* agg    = (float*)d_ws;                         // N*128 f32
    float* psum   = agg    + (size_t)N * GCN_D;           // n_stat_blk*128
    float* psumsq = psum   + (size_t)n_stat_blk * GCN_D;  // n_stat_blk*128
    float* scale  = psumsq + (size_t)n_stat_blk * GCN_D;  // 128
    float* shift  = scale  + GCN_D;                       // 128

    // 1) zero accumulator (fresh every call/replay)
    hipMemsetAsync(agg, 0, (size_t)N * GCN_D * sizeof(float), stream);

    // 2) SpMM scatter: one wave per edge, 8 waves per block
    gcn_spmm_kernel<<<(E + 7) / 8, 256, 0, stream>>>(x, esrc, edst, eval, agg, E);

    // 3) fp32 WMMA GEMM + bias -> d_out (N divisible by 16)
    gcn_gemm_bias_kernel<<<N / 16, 256, 0, stream>>>(agg, W, bias, out);

    // 4) BN statistics (deterministic two-level reduction)
    gcn_colstats_kernel<<<n_stat_blk, GCN_D, 0, stream>>>(out, psum, psumsq, N);
    gcn_bnfinalize_kernel<<<1, GCN_D, 0, stream>>>(psum, psumsq, gamma, beta,
                                                   scale, shift, n_stat_blk,
                                                   1.0f / (float)N);

    // 5) in-place affine apply
    const long nquads = (long)N * GCN_D / 4;
    gcn_bnapply_kernel<<<2048, 256, 0, stream>>>(out, scale, shift, nquads);
}